// SAModule_5970004541817
// MI455X (gfx1250) — compile-verified
//
#include <hip/hip_runtime.h>
#include <hip/hip_bf16.h>
#include <limits.h>

// ---------------- problem constants (from the reference) ----------------
#define NPTS   262144        // sum(LENGTHS)
#define LMAX   40960
#define BATCH  8
#define FEAT   64
#define CH     64
#define GRID_SZ 0.02f
#define EPSBN  1e-5f
#define HSIZE  131072        // 2^17 > max voxel id (<= 124999)
#define NPAD   65536         // B*LMAX - NPTS
#define TOTSLOT 327680.0f    // B*LMAX
#define NTILES (NPTS/16)     // 16384
#define GEMM_BLOCKS 512
#define TILES_PER_WAVE (NTILES / (GEMM_BLOCKS * 8))   // 4, exact

typedef float v2f __attribute__((ext_vector_type(2)));
typedef float v8f __attribute__((ext_vector_type(8)));

__device__ __forceinline__ int batch_of(int p) {
    int b = 0;
    b += p >= 40960;  b += p >= 77824;  b += p >= 110592; b += p >= 139264;
    b += p >= 172032; b += p >= 202752; b += p >= 237568;
    return b;
}

// ---------------- meta kernels ----------------
__global__ void k_init_minmax(int* minmax) {
    int b = threadIdx.x;
    if (b < BATCH) {
        bool pad = (b != 0);                 // only batch 0 has len == LMAX
        int mn = pad ? 50 : INT_MAX;         // padded coord 1.0 -> voxel 50
        int mx = pad ? 50 : INT_MIN;
        for (int d = 0; d < 3; ++d) { minmax[b*6+d] = mn; minmax[b*6+3+d] = mx; }
    }
}

__global__ void k_minmax(const float* __restrict__ x, int* minmax) {
    int p = blockIdx.x * blockDim.x + threadIdx.x;
    if (p >= NPTS) return;
    int b = batch_of(p);
    #pragma unroll
    for (int d = 0; d < 3; ++d) {
        int v = (int)(x[p*3+d] / GRID_SZ);
        atomicMin(&minmax[b*6+d], v);
        atomicMax(&minmax[b*6+3+d], v);
    }
}

__global__ void k_rng(const int* __restrict__ minmax, int* rngs) {
    int b = threadIdx.x;
    if (b < BATCH) {
        int r0 = minmax[b*6+3] - minmax[b*6+0];
        int r1 = minmax[b*6+4] - minmax[b*6+1];
        rngs[b*2+0] = r0;
        rngs[b*2+1] = r0 * r1;
    }
}

__global__ void k_ids(const float* __restrict__ x, const int* __restrict__ rngs,
                      int* __restrict__ ids, int* __restrict__ hist) {
    int p = blockIdx.x * blockDim.x + threadIdx.x;
    if (p >= NPTS) return;
    int b = batch_of(p);
    int v0 = (int)(x[p*3+0] / GRID_SZ);
    int v1 = (int)(x[p*3+1] / GRID_SZ);
    int v2 = (int)(x[p*3+2] / GRID_SZ);
    int id = v0 + rngs[b*2+0] * v1 + rngs[b*2+1] * v2;   // exact (< 2^17)
    ids[p] = id;
    atomicAdd(&hist[b*HSIZE + id], 1);
}

// one block per batch: occupancy prefix scan over the voxel-id space
__global__ void k_scan(const int* __restrict__ hist, int* __restrict__ rankm,
                       int* __restrict__ cnts) {
    __shared__ int sf[1024];
    int b = blockIdx.x, tid = threadIdx.x;
    int base = b * HSIZE;
    int carry = 0;
    for (int chunk = 0; chunk < HSIZE/1024; ++chunk) {
        int idx = chunk * 1024 + tid;
        int f = (hist[base + idx] > 0) ? 1 : 0;
        sf[tid] = f;
        __syncthreads();
        for (int off = 1; off < 1024; off <<= 1) {      // Hillis-Steele inclusive
            int v = (tid >= off) ? sf[tid - off] : 0;
            __syncthreads();
            sf[tid] += v;
            __syncthreads();
        }
        if (f) rankm[base + idx] = carry + sf[tid] - 1; // exclusive rank
        int total = sf[1023];
        __syncthreads();
        carry += total;
    }
    if (tid == 0) cnts[b] = carry - ((b == 0) ? 1 : 0); // full row drops its last voxel
}

__global__ void k_offsets(const int* __restrict__ cnts, int* __restrict__ offo,
                          int* __restrict__ outCounts) {
    if (threadIdx.x == 0 && blockIdx.x == 0) {
        int off = 0;
        for (int b = 0; b < BATCH; ++b) {
            offo[b] = off;
            outCounts[b] = cnts[b];
            off += cnts[b];
        }
    }
}

__global__ void k_point(const int* __restrict__ ids, const int* __restrict__ rankm,
                        const int* __restrict__ hist, const int* __restrict__ cnts,
                        const int* __restrict__ offo,
                        int* __restrict__ segslot, float* __restrict__ winv) {
    int p = blockIdx.x * blockDim.x + threadIdx.x;
    if (p >= NPTS) return;
    int b = batch_of(p);
    int id = ids[p];
    int r = rankm[b*HSIZE + id];
    if (r < cnts[b]) {
        segslot[p] = offo[b] + r;
        winv[p] = 1.0f / (float)hist[b*HSIZE + id];
    } else {
        segslot[p] = -1;
        winv[p] = 0.0f;
    }
}

__global__ void k_coords(const float* __restrict__ x, const int* __restrict__ segslot,
                         const float* __restrict__ winv, float* __restrict__ outc) {
    int p = blockIdx.x * blockDim.x + threadIdx.x;
    if (p >= NPTS) return;
    int s = segslot[p];
    if (s < 0) return;
    float wv = winv[p];
    #pragma unroll
    for (int d = 0; d < 3; ++d)
        atomicAdd(&outc[s*3 + d], x[p*3 + d] * wv);
}

// ---------------- WMMA GEMM: h = y * W^T + b ----------------
// B fragments packed so each lane's 16 v2f frags for one c0 are 32 contiguous
// floats -> compiler merges them into ds_load_b128 reads.
//   sWB[((c*2 + half)*16 + k)*2 + j] = W[c][4k + 2*half + j]
// MODE 0: accumulate per-channel sum(h), sum(h^2) for batch-norm stats
// MODE 1: apply folded BN affine + ReLU, scatter h/cnt into pooled features
template <int MODE>
__global__ void k_gemm(const float* __restrict__ y, const float* __restrict__ Wm,
                       const float* __restrict__ bias,
                       const int* __restrict__ segslot, const float* __restrict__ winv,
                       float* __restrict__ gsum, float* __restrict__ gsq,
                       const float* __restrict__ scl, const float* __restrict__ shf,
                       float* __restrict__ featsOut) {
    __shared__ float sWB[64*64];       // packed B fragments (16 KB)
    __shared__ float ssum[64], ssq[64];
    int tid = threadIdx.x;
    for (int i = tid; i < 64*64; i += 256) {
        int j    =  i        & 1;
        int k    = (i >> 1)  & 15;
        int half = (i >> 5)  & 1;
        int c    =  i >> 6;
        sWB[i] = Wm[c*64 + 4*k + 2*half + j];
    }
    if (MODE == 0 && tid < 64) { ssum[tid] = 0.0f; ssq[tid] = 0.0f; }
    __syncthreads();

    int lane = tid & 31, wave = tid >> 5;
    int half = lane >> 4, lm = lane & 15;
    int tbase = blockIdx.x * 8 + wave;

    for (int it = 0; it < TILES_PER_WAVE; ++it) {
        // Barrier each iteration: trip count is identical for all waves in the
        // block, and the fence stops LICM from hoisting the LDS B-frag reads
        // out of the loop (which previously spilled 128 VGPRs to scratch).
        __syncthreads();
        int t  = tbase + it * (GEMM_BLOCKS * 8);
        int p0 = t * 16;
        // A fragments: 16 points x 64 feats, 16 K-steps of 16x4.
        // ISA f32 A layout: lanes 0-15 hold K={0,1}, lanes 16-31 hold K={2,3}.
        const float* yr = y + (size_t)(p0 + lm) * 64 + 2 * half;
        v2f a[16];
        #pragma unroll
        for (int k = 0; k < 16; ++k) a[k] = *(const v2f*)(yr + 4 * k);

        #pragma unroll
        for (int c0i = 0; c0i < 4; ++c0i) {
            int c0 = c0i * 16;
            int c  = c0 + lm;                        // output channel for this lane
            float bv = bias[c];
            v8f acc = { bv, bv, bv, bv, bv, bv, bv, bv };   // seed C with bias
            const v2f* wrp = (const v2f*)&sWB[(c*2 + half) * 32];
            #pragma unroll
            for (int k = 0; k < 16; ++k) {
                v2f bm = wrp[k];
                acc = __builtin_amdgcn_wmma_f32_16x16x4_f32(
                        false, a[k], false, bm, (short)0, acc, false, false);
            }
            if (MODE == 0) {
                float s = 0.0f, s2 = 0.0f;
                #pragma unroll
                for (int j = 0; j < 8; ++j) { float v = acc[j]; s += v; s2 += v * v; }
                atomicAdd(&ssum[c], s);
                atomicAdd(&ssq[c], s2);
            } else {
                float sc = scl[c], sh = shf[c];
                #pragma unroll
                for (int j = 0; j < 8; ++j) {
                    int p = p0 + j + 8 * half;       // D row -> point index
                    float v = sc * acc[j] + sh;
                    v = v > 0.0f ? v : 0.0f;         // ReLU
                    int s = segslot[p];
                    if (s >= 0)
                        atomicAdd(&featsOut[(size_t)s * 64 + c], v * winv[p]);
                }
            }
        }
    }
    if (MODE == 0) {
        __syncthreads();
        if (tid < 64) { atomicAdd(&gsum[tid], ssum[tid]); atomicAdd(&gsq[tid], ssq[tid]); }
    }
}

// padding rows are constant h_pad = b - sum_f W[c,f]; NPAD copies contribute to BN stats
__global__ void k_bn_pad(const float* __restrict__ Wm, const float* __restrict__ bias,
                         float* __restrict__ gsum, float* __restrict__ gsq) {
    int c = threadIdx.x;
    if (c < CH) {
        float s = 0.0f;
        for (int f = 0; f < FEAT; ++f) s += Wm[c*FEAT + f];
        float hp = bias[c] - s;
        float np = (float)NPAD;
        atomicAdd(&gsum[c], np * hp);
        atomicAdd(&gsq[c],  np * hp * hp);
    }
}

__global__ void k_bn_final(const float* __restrict__ gsum, const float* __restrict__ gsq,
                           const float* __restrict__ gamma, const float* __restrict__ beta,
                           float* __restrict__ scl, float* __restrict__ shf) {
    int c = threadIdx.x;
    if (c < CH) {
        float mu  = gsum[c] / TOTSLOT;
        float var = gsq[c] / TOTSLOT - mu * mu;
        var = var > 0.0f ? var : 0.0f;
        float rs = rsqrtf(var + EPSBN);
        float s = gamma[c] * rs;
        scl[c] = s;
        shf[c] = beta[c] - mu * s;
    }
}

// ---------------- launcher ----------------
extern "C" void kernel_launch(void* const* d_in, const int* in_sizes, int n_in,
                              void* d_out, int out_size, void* d_ws, size_t ws_size,
                              hipStream_t stream) {
    const float* x     = (const float*)d_in[0];   // [N,3]
    const float* y     = (const float*)d_in[1];   // [N,64]
    const float* Wm    = (const float*)d_in[3];   // [64,64]
    const float* bias  = (const float*)d_in[4];   // [64]
    const float* gamma = (const float*)d_in[5];   // [64]
    const float* beta  = (const float*)d_in[6];   // [64]

    // output layout: coords[K*3] | feats[K*64] | counts[8] (int bits)
    int K = (out_size - 8) / 67;
    float* outCoords = (float*)d_out;
    float* outFeats  = outCoords + (size_t)K * 3;
    int*   outCounts = (int*)(outCoords + (size_t)K * 67);

    // workspace carve-up (~11 MB)
    char* w = (char*)d_ws;
    int*   ids     = (int*)w;                       w += sizeof(int) * NPTS;
    int*   segslot = (int*)w;                       w += sizeof(int) * NPTS;
    float* winv    = (float*)w;                     w += sizeof(float) * NPTS;
    int*   hist    = (int*)w;                       w += sizeof(int) * BATCH * HSIZE;
    int*   rankm   = (int*)w;                       w += sizeof(int) * BATCH * HSIZE;
    int*   minmax  = (int*)w;                       w += sizeof(int) * BATCH * 6;
    int*   rngs    = (int*)w;                       w += sizeof(int) * BATCH * 2;
    int*   cnts    = (int*)w;                       w += sizeof(int) * BATCH;
    int*   offo    = (int*)w;                       w += sizeof(int) * BATCH;
    float* gsum    = (float*)w;                     w += sizeof(float) * CH;
    float* gsq     = (float*)w;                     w += sizeof(float) * CH;
    float* scl     = (float*)w;                     w += sizeof(float) * CH;
    float* shf     = (float*)w;                     w += sizeof(float) * CH;

    // zero-init accumulation targets (graph-capture-safe)
    hipMemsetAsync(hist, 0, sizeof(int) * BATCH * HSIZE, stream);
    hipMemsetAsync(gsum, 0, sizeof(float) * CH * 2, stream);     // gsum + gsq
    hipMemsetAsync(d_out, 0, sizeof(float) * (size_t)out_size, stream);

    const int PB = (NPTS + 255) / 256;

    k_init_minmax<<<1, 64, 0, stream>>>(minmax);
    k_minmax<<<PB, 256, 0, stream>>>(x, minmax);
    k_rng<<<1, 8, 0, stream>>>(minmax, rngs);
    k_ids<<<PB, 256, 0, stream>>>(x, rngs, ids, hist);
    k_scan<<<BATCH, 1024, 0, stream>>>(hist, rankm, cnts);
    k_offsets<<<1, 1, 0, stream>>>(cnts, offo, outCounts);
    k_point<<<PB, 256, 0, stream>>>(ids, rankm, hist, cnts, offo, segslot, winv);
    k_coords<<<PB, 256, 0, stream>>>(x, segslot, winv, outCoords);

    k_gemm<0><<<GEMM_BLOCKS, 256, 0, stream>>>(y, Wm, bias, segslot, winv,
                                               gsum, gsq, scl, shf, outFeats);
    k_bn_pad<<<1, 64, 0, stream>>>(Wm, bias, gsum, gsq);
    k_bn_final<<<1, 64, 0, stream>>>(gsum, gsq, gamma, beta, scl, shf);
    k_gemm<1><<<GEMM_BLOCKS, 256, 0, stream>>>(y, Wm, bias, segslot, winv,
                                               gsum, gsq, scl, shf, outFeats);
}